// PPOLinearModel_46875273068984
// MI455X (gfx1250) — compile-verified
//
#include <hip/hip_runtime.h>

#define H 128
#define HOPS 3

typedef __attribute__((ext_vector_type(2))) float v2f;
typedef __attribute__((ext_vector_type(8))) float v8f;
typedef __attribute__((ext_vector_type(4))) unsigned tdm4u;
typedef __attribute__((ext_vector_type(8))) int tdm8i;
typedef __attribute__((ext_vector_type(4))) int tdm4i;

// ---------------------------------------------------------------- TDM staging
// Load a 16x128 fp32 tile (rows clamped by rowsAvail -> OOB rows read as zero)
// into LDS with 4-DWORD padding every 128 DWORDs (row stride 132 floats),
// matching the WMMA fragment-read layout. One wave issues this; others wait at
// the block barrier.
#if __has_builtin(__builtin_amdgcn_tensor_load_to_lds)
#define USE_TDM_STAGING 1
__device__ __forceinline__ void tdm_load_tile16x128(float* ldsDst, const float* gsrc,
                                                    int rowsAvail) {
  unsigned long long ga = (unsigned long long)(uintptr_t)gsrc;
  unsigned ldsOff = (unsigned)(uintptr_t)ldsDst;  // low 32 bits = LDS byte offset
  tdm4u g0;
  g0.x = 1u;                                        // count=1, user descriptor
  g0.y = ldsOff;                                    // lds_addr
  g0.z = (unsigned)(ga & 0xffffffffu);              // global_addr[31:0]
  g0.w = (unsigned)((ga >> 32) & 0x1ffffffu)        // global_addr[56:32]
         | (2u << 30);                              // type=2 ("image")
  unsigned td0 = (unsigned)H;                       // tensor_dim0 = 128
  unsigned td1 = (unsigned)rowsAvail;               // tensor_dim1 (OOB -> zeros)
  tdm8i g1;
  // data_size=4B (2<<16) | pad_enable (1<<20) | pad_interval=128dw (6<<22)
  // | pad_amount=4dw (3<<25)
  g1[0] = (int)0x07920000u;
  g1[1] = (int)((td0 & 0xffffu) << 16);                               // dim0 lo
  g1[2] = (int)(((td0 >> 16) & 0xffffu) | ((td1 & 0xffffu) << 16));   // dim0 hi | dim1 lo
  g1[3] = (int)(((td1 >> 16) & 0xffffu) | ((unsigned)H << 16));       // dim1 hi | tile_dim0=128
  g1[4] = 16;                                                         // tile_dim1=16, tile_dim2=0
  g1[5] = H;                                                          // tensor_dim0_stride=128
  g1[6] = 0;
  g1[7] = 0;
  tdm4i gz = {0, 0, 0, 0};
#if defined(__clang_major__) && __clang_major__ >= 23
  tdm8i z8 = {0, 0, 0, 0, 0, 0, 0, 0};
  __builtin_amdgcn_tensor_load_to_lds(g0, g1, gz, gz, z8, 0);
#else
  __builtin_amdgcn_tensor_load_to_lds(g0, g1, gz, gz, 0);
#endif
}
#else
#define USE_TDM_STAGING 0
#endif

// ---------------------------------------------------------------- utilities
__global__ void zero_i32_kernel(int* __restrict__ p, int n) {
  int i = blockIdx.x * blockDim.x + threadIdx.x;
  if (i < n) p[i] = 0;
}

__global__ void hist_kernel(const int* __restrict__ row, int* __restrict__ deg, int e) {
  int i = blockIdx.x * blockDim.x + threadIdx.x;
  if (i < e) atomicAdd(&deg[row[i]], 1);
}

__global__ __launch_bounds__(256) void scan1_kernel(const int* __restrict__ deg,
                                                    int* __restrict__ ptr,
                                                    int* __restrict__ bsum, int n) {
  __shared__ int wsum[9];
  int lane = threadIdx.x & 31;
  int wv   = threadIdx.x >> 5;
  int base = blockIdx.x * 1024 + threadIdx.x * 4;
  int e0 = (base + 0 < n) ? deg[base + 0] : 0;
  int e1 = (base + 1 < n) ? deg[base + 1] : 0;
  int e2 = (base + 2 < n) ? deg[base + 2] : 0;
  int e3 = (base + 3 < n) ? deg[base + 3] : 0;
  int s = e0 + e1 + e2 + e3;
  int inc = s;
#pragma unroll
  for (int d = 1; d < 32; d <<= 1) {
    int t = __shfl_up(inc, d, 32);
    if (lane >= d) inc += t;
  }
  if (lane == 31) wsum[wv] = inc;
  __syncthreads();
  if (threadIdx.x == 0) {
    int run = 0;
    for (int i = 0; i < 8; ++i) { int t = wsum[i]; wsum[i] = run; run += t; }
    wsum[8] = run;
  }
  __syncthreads();
  int exc = wsum[wv] + inc - s;
  if (base + 0 < n) ptr[base + 0] = exc; exc += e0;
  if (base + 1 < n) ptr[base + 1] = exc; exc += e1;
  if (base + 2 < n) ptr[base + 2] = exc; exc += e2;
  if (base + 3 < n) ptr[base + 3] = exc;
  if (threadIdx.x == 0) bsum[blockIdx.x] = wsum[8];
}

__global__ void scan2_kernel(const int* __restrict__ bsum, int* __restrict__ boff, int nb) {
  if (threadIdx.x == 0 && blockIdx.x == 0) {
    int run = 0;
    for (int i = 0; i < nb; ++i) { int t = bsum[i]; boff[i] = run; run += t; }
    boff[nb] = run;
  }
}

__global__ void scan3_kernel(int* __restrict__ ptr, const int* __restrict__ deg,
                             const int* __restrict__ boff, float* __restrict__ degf,
                             int n, int nb) {
  int i = blockIdx.x * blockDim.x + threadIdx.x;
  if (i < n) {
    ptr[i] += boff[i >> 10];
    degf[i] = (float)deg[i];
  }
  if (i == 0) ptr[n] = boff[nb];
}

__global__ void fill_kernel(const int* __restrict__ ei, const int* __restrict__ ptr,
                            int* __restrict__ cur, int* __restrict__ cols, int e) {
  int i = blockIdx.x * blockDim.x + threadIdx.x;
  if (i < e) {
    int r   = ei[i];
    int pos = ptr[r] + atomicAdd(&cur[r], 1);
    cols[pos] = ei[e + i];
  }
}

// v3[l][o] = sum_k relu(W4[l][k]) * W3[l][o][k]
__global__ void v3_kernel(const float* __restrict__ W3, const float* __restrict__ W4,
                          float* __restrict__ v3) {
  int l = blockIdx.x, o = threadIdx.x;
  float acc = 0.f;
  const float* w3row = W3 + ((size_t)l * H + o) * H;
  const float* w4v   = W4 + (size_t)l * H;
  for (int k = 0; k < H; ++k) acc += fmaxf(w4v[k], 0.f) * w3row[k];
  v3[l * H + o] = acc;
}

// ----------------------------------------------- neighbor aggregation (L2-hit)
__global__ __launch_bounds__(256) void agg_kernel(const float* __restrict__ h,
                                                  const int* __restrict__ ptr,
                                                  const int* __restrict__ cols,
                                                  float* __restrict__ agg, int n) {
  int gw   = (blockIdx.x * blockDim.x + threadIdx.x) >> 5;
  int lane = threadIdx.x & 31;
  if (gw >= n) return;
  int beg = ptr[gw], end = ptr[gw + 1];
  float4 acc = {0.f, 0.f, 0.f, 0.f};
  for (int j0 = beg; j0 < end; j0 += 32) {
    int cnt = end - j0; if (cnt > 32) cnt = 32;
    int idx = (lane < cnt) ? cols[j0 + lane] : 0;
    for (int t = 0; t < cnt; ++t) {
      int nb = __shfl(idx, t, 32);
      const float4 hv = *(const float4*)&h[(size_t)nb * H + lane * 4];
      acc.x += hv.x; acc.y += hv.y; acc.z += hv.z; acc.w += hv.w;
    }
  }
  *(float4*)&agg[(size_t)gw * H + lane * 4] = acc;
}

// ------------------------------------------- fused hop GEMM (fp32 WMMA, exact)
// h_new = relu(x@W1.T + agg@W2.T + deg*v3)
// 256 thr = 8 waves; 16 rows/block; wave -> 16-col slab; dual WMMA accumulators
template <bool USEAGG>
__global__ __launch_bounds__(256) void hop_kernel(const float* __restrict__ X,
                                                  const float* __restrict__ AGG,
                                                  const float* __restrict__ W1l,
                                                  const float* __restrict__ W2l,
                                                  const float* __restrict__ v3l,
                                                  const float* __restrict__ degf,
                                                  float* __restrict__ Hout, int n) {
  __shared__ float xs[16 * 132];
  __shared__ float as_[16 * 132];
  int rowBase = blockIdx.x * 16;

#if USE_TDM_STAGING
  if (threadIdx.x < 32) {
    tdm_load_tile16x128(xs, X + (size_t)rowBase * H, n - rowBase);
    if (USEAGG) tdm_load_tile16x128(as_, AGG + (size_t)rowBase * H, n - rowBase);
    __builtin_amdgcn_s_wait_tensorcnt(0);
  }
#else
  for (int i = threadIdx.x; i < 16 * H; i += 256) {
    int r = i >> 7, c = i & 127;
    int rr = rowBase + r;
    xs[r * 132 + c] = (rr < n) ? X[(size_t)rr * H + c] : 0.f;
    if (USEAGG) as_[r * 132 + c] = (rr < n) ? AGG[(size_t)rr * H + c] : 0.f;
  }
#endif
  __syncthreads();

  int lane  = threadIdx.x & 31;
  int wv    = threadIdx.x >> 5;
  int nn    = lane & 15;
  int o     = wv * 16 + nn;
  int mhalf = (lane < 16) ? 0 : 8;
  int koff  = (lane < 16) ? 0 : 2;
  float vv  = v3l[o];

  v8f c1, c2;
#pragma unroll
  for (int i = 0; i < 8; ++i) {
    int ri = rowBase + i + mhalf;
    c1[i] = ((ri < n) ? degf[ri] : 0.f) * vv;
    c2[i] = 0.f;
  }

  const float* w1row = W1l + (size_t)o * H;
  const float* w2row = W2l + (size_t)o * H;
#pragma unroll
  for (int k = 0; k < H; k += 4) {
    int kA = k + koff;
    v2f a; a.x = xs[nn * 132 + kA]; a.y = xs[nn * 132 + kA + 1];
    v2f b = *(const v2f*)&w1row[kA];
    c1 = __builtin_amdgcn_wmma_f32_16x16x4_f32(false, a, false, b, (short)0, c1, false, false);
    if (USEAGG) {
      v2f a2; a2.x = as_[nn * 132 + kA]; a2.y = as_[nn * 132 + kA + 1];
      v2f b2 = *(const v2f*)&w2row[kA];
      c2 = __builtin_amdgcn_wmma_f32_16x16x4_f32(false, a2, false, b2, (short)0, c2, false, false);
    }
  }
#pragma unroll
  for (int i = 0; i < 8; ++i) {
    int row = rowBase + i + mhalf;
    float r = USEAGG ? (c1[i] + c2[i]) : c1[i];
    if (row < n) Hout[(size_t)row * H + o] = fmaxf(r, 0.f);
  }
}

// ------------------------------------------- actor layer 1: relu(h@aW1.T + b)
__global__ __launch_bounds__(128) void actor1_kernel(const float* __restrict__ Hin,
                                                     const float* __restrict__ aW1,
                                                     const float* __restrict__ ab1,
                                                     float* __restrict__ A1, int n) {
  __shared__ float hs[16 * 132];
  int rowBase = blockIdx.x * 16;
#if USE_TDM_STAGING
  if (threadIdx.x < 32) {
    tdm_load_tile16x128(hs, Hin + (size_t)rowBase * H, n - rowBase);
    __builtin_amdgcn_s_wait_tensorcnt(0);
  }
#else
  for (int i = threadIdx.x; i < 16 * H; i += 128) {
    int r = i >> 7, cc = i & 127;
    int rr = rowBase + r;
    hs[r * 132 + cc] = (rr < n) ? Hin[(size_t)rr * H + cc] : 0.f;
  }
#endif
  __syncthreads();
  int lane  = threadIdx.x & 31;
  int wv    = threadIdx.x >> 5;
  int nn    = lane & 15;
  int o     = wv * 16 + nn;
  int mhalf = (lane < 16) ? 0 : 8;
  int koff  = (lane < 16) ? 0 : 2;
  float bias = ab1[o];
  v8f c;
#pragma unroll
  for (int i = 0; i < 8; ++i) c[i] = bias;
  const float* wrow = aW1 + (size_t)o * H;
#pragma unroll
  for (int k = 0; k < H; k += 4) {
    int kA = k + koff;
    v2f a; a.x = hs[nn * 132 + kA]; a.y = hs[nn * 132 + kA + 1];
    v2f b = *(const v2f*)&wrow[kA];
    c = __builtin_amdgcn_wmma_f32_16x16x4_f32(false, a, false, b, (short)0, c, false, false);
  }
#pragma unroll
  for (int i = 0; i < 8; ++i) {
    int row = rowBase + i + mhalf;
    if (row < n) A1[(size_t)row * 64 + o] = fmaxf(c[i], 0.f);
  }
}

// -------------------------- actor layers 2+3: wave-per-node (64->32->1)
__global__ __launch_bounds__(256) void actor23_kernel(const float* __restrict__ A1,
                                                      const float* __restrict__ aW2,
                                                      const float* __restrict__ ab2,
                                                      const float* __restrict__ aW3,
                                                      const float* __restrict__ ab3,
                                                      float* __restrict__ logits, int n) {
  __shared__ float w2s[32 * 65];
  for (int i = threadIdx.x; i < 32 * 64; i += 256)
    w2s[(i >> 6) * 65 + (i & 63)] = aW2[i];
  __syncthreads();
  int lane = threadIdx.x & 31;
  int node = blockIdx.x * 8 + (threadIdx.x >> 5);
  if (node >= n) return;
  float alo = A1[(size_t)node * 64 + lane];
  float ahi = A1[(size_t)node * 64 + 32 + lane];
  float acc = ab2[lane];
#pragma unroll 8
  for (int k = 0; k < 32; ++k) {
    acc += __shfl(alo, k, 32) * w2s[lane * 65 + k];
    acc += __shfl(ahi, k, 32) * w2s[lane * 65 + 32 + k];
  }
  acc = fmaxf(acc, 0.f) * aW3[lane];
#pragma unroll
  for (int d = 16; d > 0; d >>= 1) acc += __shfl_xor(acc, d, 32);
  if (lane == 0) logits[node] = acc + ab3[0];
}

// ----------------------------------------------- pooling partials (sum & max)
__global__ __launch_bounds__(128) void pool_kernel(const float* __restrict__ h,
                                                   float* __restrict__ psum,
                                                   float* __restrict__ pmax, int n) {
  int f = threadIdx.x;
  float s = 0.f, m = -3.402823466e38f;
  for (int r = blockIdx.x; r < n; r += gridDim.x) {
    float v = h[(size_t)r * H + f];
    s += v; m = fmaxf(m, v);
  }
  psum[blockIdx.x * H + f] = s;
  pmax[blockIdx.x * H + f] = m;
}

// ----------------------------------------------- critic: reduce + MLP
__global__ __launch_bounds__(256) void critic_kernel(const float* __restrict__ psum,
                                                     const float* __restrict__ pmax,
                                                     int nblk, int n,
                                                     const float* __restrict__ cW1,
                                                     const float* __restrict__ cb1,
                                                     const float* __restrict__ cW2,
                                                     const float* __restrict__ cb2,
                                                     const float* __restrict__ cW3,
                                                     const float* __restrict__ cb3,
                                                     float* __restrict__ out) {
  __shared__ float g[3 * H];
  __shared__ float c1[256];
  __shared__ float c2[128];
  __shared__ float red[128];
  int t = threadIdx.x;
  if (t < H) {
    float s = 0.f, m = -3.402823466e38f;
    for (int b = 0; b < nblk; ++b) {
      s += psum[b * H + t];
      m = fmaxf(m, pmax[b * H + t]);
    }
    g[t]         = s / (float)n;
    g[H + t]     = m;
    g[2 * H + t] = s;
  }
  __syncthreads();
  {
    float acc = cb1[t];
    const float* wrow = cW1 + (size_t)t * (3 * H);
    for (int k = 0; k < 3 * H; ++k) acc += g[k] * wrow[k];
    c1[t] = fmaxf(acc, 0.f);
  }
  __syncthreads();
  if (t < 128) {
    float acc = cb2[t];
    const float* wrow = cW2 + (size_t)t * 256;
    for (int k = 0; k < 256; ++k) acc += c1[k] * wrow[k];
    c2[t] = fmaxf(acc, 0.f);
  }
  __syncthreads();
  if (t < 128) red[t] = c2[t] * cW3[t];
  __syncthreads();
  for (int s2 = 64; s2 > 0; s2 >>= 1) {
    if (t < s2) red[t] += red[t + s2];
    __syncthreads();
  }
  if (t == 0) out[0] = red[0] + cb3[0];
}

// ================================================================ launcher
extern "C" void kernel_launch(void* const* d_in, const int* in_sizes, int n_in,
                              void* d_out, int out_size, void* d_ws, size_t ws_size,
                              hipStream_t stream) {
  (void)n_in; (void)out_size; (void)ws_size;
  const float* x   = (const float*)d_in[0];
  const int*   ei  = (const int*)d_in[1];
  const float* W1  = (const float*)d_in[2];
  const float* W2  = (const float*)d_in[3];
  const float* W3  = (const float*)d_in[4];
  const float* W4  = (const float*)d_in[5];
  const float* aW1 = (const float*)d_in[6];
  const float* ab1 = (const float*)d_in[7];
  const float* aW2 = (const float*)d_in[8];
  const float* ab2 = (const float*)d_in[9];
  const float* aW3 = (const float*)d_in[10];
  const float* ab3 = (const float*)d_in[11];
  const float* cW1 = (const float*)d_in[12];
  const float* cb1 = (const float*)d_in[13];
  const float* cW2 = (const float*)d_in[14];
  const float* cb2 = (const float*)d_in[15];
  const float* cW3 = (const float*)d_in[16];
  const float* cb3 = (const float*)d_in[17];

  const int n = in_sizes[0] / H;   // 50000
  const int e = in_sizes[1] / 2;   // 1600000

  char* wsb = (char*)d_ws;
  size_t off = 0;
  auto carve = [&](size_t bytes) -> void* {
    void* p = wsb + off;
    off += (bytes + (size_t)255) & ~(size_t)255;
    return p;
  };
  int*   deg  = (int*)carve((size_t)n * 4);
  int*   cur  = (int*)carve((size_t)n * 4);
  int*   ptr  = (int*)carve((size_t)(n + 1) * 4);
  int*   bsum = (int*)carve(256);
  int*   boff = (int*)carve(512);
  float* degf = (float*)carve((size_t)n * 4);
  int*   ccol = (int*)carve((size_t)e * 4);
  float* v3   = (float*)carve((size_t)HOPS * H * 4);
  float* h    = (float*)carve((size_t)n * H * 4);
  float* agg  = (float*)carve((size_t)n * H * 4);  // reused as a1 (needs n*64)
  float* psum = (float*)carve((size_t)128 * H * 4);
  float* pmax = (float*)carve((size_t)128 * H * 4);
  float* a1   = agg;

  float* logits = (float*)d_out;
  float* value  = (float*)d_out + n;

  const int nb   = (n + 1023) / 1024;
  const int nr16 = (n + 15) / 16;

  // ---- CSR build
  zero_i32_kernel<<<(n + 255) / 256, 256, 0, stream>>>(deg, n);
  zero_i32_kernel<<<(n + 255) / 256, 256, 0, stream>>>(cur, n);
  hist_kernel<<<(e + 255) / 256, 256, 0, stream>>>(ei, deg, e);
  scan1_kernel<<<nb, 256, 0, stream>>>(deg, ptr, bsum, n);
  scan2_kernel<<<1, 32, 0, stream>>>(bsum, boff, nb);
  scan3_kernel<<<(n + 255) / 256, 256, 0, stream>>>(ptr, deg, boff, degf, n, nb);
  fill_kernel<<<(e + 255) / 256, 256, 0, stream>>>(ei, ptr, cur, ccol, e);

  // ---- per-hop edge-feature vector v3[l] = relu(W4[l]) @ W3[l].T
  v3_kernel<<<HOPS, H, 0, stream>>>(W3, W4, v3);

  // ---- hop 0 (h==0 -> skip aggregation entirely)
  hop_kernel<false><<<nr16, 256, 0, stream>>>(x, agg, W1, W2, v3, degf, h, n);
  // ---- hops 1..2
  for (int l = 1; l < HOPS; ++l) {
    agg_kernel<<<(n + 7) / 8, 256, 0, stream>>>(h, ptr, ccol, agg, n);
    hop_kernel<true><<<nr16, 256, 0, stream>>>(x, agg,
                                               W1 + (size_t)l * H * H,
                                               W2 + (size_t)l * H * H,
                                               v3 + (size_t)l * H, degf, h, n);
  }

  // ---- critic pooling partials (h resident in L2)
  pool_kernel<<<128, 128, 0, stream>>>(h, psum, pmax, n);
  // ---- actor
  actor1_kernel<<<nr16, 128, 0, stream>>>(h, aW1, ab1, a1, n);
  actor23_kernel<<<(n + 7) / 8, 256, 0, stream>>>(a1, aW2, ab2, aW3, ab3, logits, n);
  // ---- critic MLP
  critic_kernel<<<1, 256, 0, stream>>>(psum, pmax, 128, n,
                                       cW1, cb1, cW2, cb2, cW3, cb3, value);
}